// EnhanceDiversityFeatureExtracition_69638599737822
// MI455X (gfx1250) — compile-verified
//
#include <hip/hip_runtime.h>
#include <hip/hip_bf16.h>

typedef __attribute__((ext_vector_type(2))) float v2f;
typedef __attribute__((ext_vector_type(8))) float v8f;

#define VOCAB 50257
#define NROWS 8192
#define NB (512 * 512)          // number of (o,c) 3x3 tiles
#define CHUNKS ((NB + 4) / 5)   // 5 tiles per WMMA
#define REG_BLOCKS 128
#define NEG_HUGE (-3.402823466e38f)

// ---------------------------------------------------------------------------
// Kernel 1: one-pass online logsumexp cross-entropy. One block per row.
// Branchless update: exactly one v_exp_f32 per element, no exec divergence.
// ---------------------------------------------------------------------------
__device__ __forceinline__ void online_update(float& m, float& s, float x) {
    float d = x - m;
    float e = __expf(-fabsf(d));     // exp(min(m-x, x-m)) : always the small side
    bool gt = d > 0.f;
    s = gt ? __fmaf_rn(s, e, 1.0f) : (s + e);
    m = gt ? x : m;
}

__device__ __forceinline__ void combine_ms(float& m, float& s, float mo, float so) {
    float d = mo - m;
    float e = __expf(-fabsf(d));
    bool gt = d > 0.f;
    s = gt ? __fmaf_rn(s, e, so) : __fmaf_rn(so, e, s);
    m = gt ? mo : m;
}

__global__ void ce_rows_kernel(const float* __restrict__ logits,
                               const int* __restrict__ labels,
                               float* __restrict__ nll) {
    const int row = blockIdx.x;
    const int tid = threadIdx.x;
    const int nth = blockDim.x;
    const float* rowp = logits + (size_t)row * VOCAB;

    // Four independent online accumulators (one per float4 component).
    float m0 = NEG_HUGE, m1 = NEG_HUGE, m2 = NEG_HUGE, m3 = NEG_HUGE;
    float s0 = 0.f, s1 = 0.f, s2 = 0.f, s3 = 0.f;

    // Align to 16 bytes for b128 streaming loads (rows are only dword-aligned).
    int mis = (int)((((uintptr_t)rowp) >> 2) & 3);   // dwords past 16B boundary
    int head = (4 - mis) & 3;

    for (int i = tid; i < head; i += nth) online_update(m0, s0, rowp[i]);

    const int nvec = (VOCAB - head) >> 2;
    const float4* vp = reinterpret_cast<const float4*>(rowp + head);
    for (int i = tid; i < nvec; i += nth) {
        float4 v = vp[i];                              // global_load_b128
        online_update(m0, s0, v.x);
        online_update(m1, s1, v.y);
        online_update(m2, s2, v.z);
        online_update(m3, s3, v.w);
    }
    for (int i = head + (nvec << 2) + tid; i < VOCAB; i += nth)
        online_update(m0, s0, rowp[i]);

    // Merge the four accumulators.
    combine_ms(m0, s0, m1, s1);
    combine_ms(m2, s2, m3, s3);
    combine_ms(m0, s0, m2, s2);

    // Block reduction over (m, s) pairs.
    __shared__ float sm[256];
    __shared__ float ss[256];
    sm[tid] = m0;
    ss[tid] = s0;
    __syncthreads();
    for (int off = nth >> 1; off > 0; off >>= 1) {
        if (tid < off) {
            float mm = sm[tid], sv = ss[tid];
            combine_ms(mm, sv, sm[tid + off], ss[tid + off]);
            sm[tid] = mm;
            ss[tid] = sv;
        }
        __syncthreads();
    }
    if (tid == 0) {
        float lse = sm[0] + __logf(ss[0]);
        int lab = labels[row];
        nll[row] = lse - rowp[lab];
    }
}

// ---------------------------------------------------------------------------
// Kernel 2: batched 3x3 gram via V_WMMA_F32_16X16X4_F32 (5 tiles / WMMA)
// plus row/col sum-of-squares partials. One 15-float partial per block:
// [0..8] dots, [9..11] sum_sq rows (i), [12..14] sum_sq cols (k).
// Inner loop is branch-free: clamped addresses + 0/1 value masks.
// ---------------------------------------------------------------------------
__global__ void reg_partials_kernel(const float* __restrict__ W,
                                    float* __restrict__ partials) {
    __shared__ float sDots[8][9];
    __shared__ float sNorm[8][6];

    const int tid = threadIdx.x;
    const int lane = tid & 31;
    const int wid = tid >> 5;                                  // wave in block
    const int gwave = blockIdx.x * (blockDim.x >> 5) + wid;
    const int nwaves = gridDim.x * (blockDim.x >> 5);
    const int halfSel = lane >> 4;                             // 0: lanes 0-15
    const int l = lane & 15;

    // K slots per ISA A-layout: VGPR0 holds K=0 (lanes 0-15) / K=2 (16-31),
    // VGPR1 holds K=1 / K=3. K=3 is zero padding (K dim is only 3).
    const int k0 = halfSel ? 2 : 0;
    const int k1 = halfSel ? 3 : 1;
    const int k1c = (k1 < 3) ? k1 : 2;     // clamped index for safe loads
    const int bi0 = l / 3;                 // tile within 5-tile chunk
    const int bi = (bi0 < 5) ? bi0 : 4;    // clamp for lane l==15
    const int ri = l - 3 * ((l / 3));      // row (A) == col (B) within tile
    const bool laneValid = (l < 15);
    const bool k1Valid = (k1 < 3);

    // Loop-invariant element offsets within a 3x3 tile.
    const int idxA0 = ri * 3 + k0;
    const int idxA1 = ri * 3 + k1c;
    const int idxB0 = k0 * 3 + ri;
    const int idxB1 = k1c * 3 + ri;

    v8f acc = {0.f, 0.f, 0.f, 0.f, 0.f, 0.f, 0.f, 0.f};

    for (int chunk = gwave; chunk < CHUNKS; chunk += nwaves) {
        const int bb = chunk * 5 + bi;
        const int bc = (bb < NB) ? bb : (NB - 1);              // clamp address
        const float mk0 = (laneValid && bb < NB) ? 1.f : 0.f;  // value mask
        const float mk1 = k1Valid ? mk0 : 0.f;
        const float* t = W + (size_t)bc * 9;
        // Four unconditional loads -> one clause, one wait, then mask+WMMA.
        float a0 = t[idxA0] * mk0;
        float a1 = t[idxA1] * mk1;
        float b0 = t[idxB0] * mk0;
        float b1 = t[idxB1] * mk1;
        v2f av = {a0, a1};
        v2f bv = {b0, b1};
        acc = __builtin_amdgcn_wmma_f32_16x16x4_f32(
            /*neg_a=*/false, av, /*neg_b=*/false, bv,
            /*c_mod=*/(short)0, acc, /*reuse_a=*/false, /*reuse_b=*/false);
    }

    // Extract the 5 diagonal 3x3 blocks of C and sum them.
    // C layout: element (M,N) lives in VGPR (M%8), lane (M>=8?16:0)+N.
    float dp[9];
#pragma unroll
    for (int i = 0; i < 3; ++i) {
#pragma unroll
        for (int j = 0; j < 3; ++j) {
            float sum = 0.f;
#pragma unroll
            for (int b = 0; b < 5; ++b) {
                const int M = 3 * b + i;
                const int N = 3 * b + j;
                float v = acc[M & 7];
                int src = ((M >= 8) ? 16 : 0) + N;
                sum += __shfl(v, src, 32);
            }
            dp[3 * i + j] = sum;
        }
    }
    if (lane == 0) {
#pragma unroll
        for (int q = 0; q < 9; ++q) sDots[wid][q] = dp[q];
    }

    // Sum-of-squares partials: one full 3x3 tile per thread iteration so all
    // accumulator indices are compile-time constants.
    float nr0 = 0.f, nr1 = 0.f, nr2 = 0.f, nc0 = 0.f, nc1 = 0.f, nc2 = 0.f;
    const int gtid = blockIdx.x * blockDim.x + tid;
    const int gstride = gridDim.x * blockDim.x;
    for (int b = gtid; b < NB; b += gstride) {
        const float* t = W + (size_t)b * 9;
        float t0 = t[0], t1 = t[1], t2 = t[2];
        float t3 = t[3], t4 = t[4], t5 = t[5];
        float t6 = t[6], t7 = t[7], t8 = t[8];
        nr0 += t0 * t0 + t1 * t1 + t2 * t2;
        nr1 += t3 * t3 + t4 * t4 + t5 * t5;
        nr2 += t6 * t6 + t7 * t7 + t8 * t8;
        nc0 += t0 * t0 + t3 * t3 + t6 * t6;
        nc1 += t1 * t1 + t4 * t4 + t7 * t7;
        nc2 += t2 * t2 + t5 * t5 + t8 * t8;
    }
#pragma unroll
    for (int off = 16; off > 0; off >>= 1) {
        nr0 += __shfl_xor(nr0, off, 32);
        nr1 += __shfl_xor(nr1, off, 32);
        nr2 += __shfl_xor(nr2, off, 32);
        nc0 += __shfl_xor(nc0, off, 32);
        nc1 += __shfl_xor(nc1, off, 32);
        nc2 += __shfl_xor(nc2, off, 32);
    }
    if (lane == 0) {
        sNorm[wid][0] = nr0; sNorm[wid][1] = nr1; sNorm[wid][2] = nr2;
        sNorm[wid][3] = nc0; sNorm[wid][4] = nc1; sNorm[wid][5] = nc2;
    }
    __syncthreads();

    // Fixed-order block combine (deterministic), one 15-float record per block.
    if (tid == 0) {
        float* rec = partials + (size_t)blockIdx.x * 15;
#pragma unroll
        for (int q = 0; q < 9; ++q) {
            float a = 0.f;
            for (int w = 0; w < 8; ++w) a += sDots[w][q];
            rec[q] = a;
        }
#pragma unroll
        for (int q = 0; q < 6; ++q) {
            float a = 0.f;
            for (int w = 0; w < 8; ++w) a += sNorm[w][q];
            rec[9 + q] = a;
        }
    }
}

// ---------------------------------------------------------------------------
// Kernel 3: final deterministic reduction + similarity mask + output scalar.
// ---------------------------------------------------------------------------
__global__ void final_kernel(const float* __restrict__ nll,
                             const float* __restrict__ partials,
                             float* __restrict__ out) {
    __shared__ float red[256];
    __shared__ float agg[15];
    const int tid = threadIdx.x;

    float s = 0.f;
    for (int i = tid; i < NROWS; i += 256) s += nll[i];
    red[tid] = s;
    __syncthreads();
    for (int off = 128; off > 0; off >>= 1) {
        if (tid < off) red[tid] += red[tid + off];
        __syncthreads();
    }

    if (tid < 15) {
        float a = 0.f;
        for (int b = 0; b < REG_BLOCKS; ++b) a += partials[(size_t)b * 15 + tid];
        agg[tid] = a;
    }
    __syncthreads();

    if (tid == 0) {
        float mean_nll = red[0] / (float)NROWS;
        float nr[3], nc[3];
#pragma unroll
        for (int q = 0; q < 3; ++q) {
            nr[q] = sqrtf(agg[9 + q]);
            nc[q] = sqrtf(agg[12 + q]);
        }
        float reg = 0.f;
#pragma unroll
        for (int i = 0; i < 3; ++i) {
#pragma unroll
            for (int j = 0; j < 3; ++j) {
                if (i != j) {
                    float sim = agg[3 * i + j] / (nr[i] * nc[j]);
                    if (sim > 0.2f && sim <= 1.0f) reg += sim;
                }
            }
        }
        out[0] = mean_nll + 0.0005f * reg;
    }
}

// ---------------------------------------------------------------------------
extern "C" void kernel_launch(void* const* d_in, const int* in_sizes, int n_in,
                              void* d_out, int out_size, void* d_ws, size_t ws_size,
                              hipStream_t stream) {
    const float* logits = (const float*)d_in[0];
    const int* labels = (const int*)d_in[1];
    const float* conv_w = (const float*)d_in[2];
    float* out = (float*)d_out;

    float* ws = (float*)d_ws;
    float* nll = ws;                    // NROWS floats
    float* partials = ws + NROWS;       // REG_BLOCKS * 15 floats

    ce_rows_kernel<<<NROWS, 256, 0, stream>>>(logits, labels, nll);
    reg_partials_kernel<<<REG_BLOCKS, 256, 0, stream>>>(conv_w, partials);
    final_kernel<<<1, 256, 0, stream>>>(nll, partials, out);
}